// Gnn_KRNet_32942399160593
// MI455X (gfx1250) — compile-verified
//
#include <hip/hip_runtime.h>
#include <math.h>

typedef __attribute__((ext_vector_type(2))) float v2f;
typedef __attribute__((ext_vector_type(8))) float v8f;

#define NN   1200000
#define EE   12000000
#define DIN  10
#define DH   20
#define DE   5
#define BB   200000
#define DR   30
#define DOUT 60
#define DX1  7

// sin(0..9) as fp32 constants (positional encoding, d_model=1 -> only sin row)
__device__ __constant__ float PO_SIN[12] = {
    0.0f,            0.8414709848f,  0.9092974268f,  0.1411200081f,
   -0.7568024953f,  -0.9589242747f, -0.2794154982f,  0.6569865987f,
    0.9893582466f,   0.4121184852f,  0.0f,           0.0f};

// ---- CDNA5 fp32 WMMA: D = A(16x4) * B(4x16) + C(16x16), full fp32 ----
__device__ __forceinline__ v8f wmma4(v2f a, v2f b, v8f c) {
  return __builtin_amdgcn_wmma_f32_16x16x4_f32(false, a, false, b, (short)0, c,
                                               false, false);
}

__device__ __forceinline__ float gelu_f(float v) {
  return 0.5f * v * (1.0f + erff(v * 0.7071067811865476f));
}

// ---------------- degree / dinv ----------------
__global__ void k_init_deg(float* deg) {
  int i = blockIdx.x * blockDim.x + threadIdx.x;
  if (i < NN) deg[i] = 1.0f;
}

__global__ void k_degree(const int* __restrict__ dst, float* deg) {
  int e = blockIdx.x * blockDim.x + threadIdx.x;
  if (e < EE) atomicAdd(&deg[dst[e]], 1.0f);
}

__global__ void k_dinv(float* dinv) {
  int i = blockIdx.x * blockDim.x + threadIdx.x;
  if (i < NN) dinv[i] = rsqrtf(dinv[i]);
}

// ---- h1pre = (x + po) @ W1  [N,10]x[10,20]; also agg = h1pre * dinv^2 ----
__global__ void k_gemm_w1(const float* __restrict__ x, const float* __restrict__ W1,
                          const float* __restrict__ dinv,
                          float* __restrict__ h1pre, float* __restrict__ agg) {
  int wid = (int)((blockIdx.x * blockDim.x + threadIdx.x) >> 5);
  if (wid >= NN / 16) return;                 // wave-uniform
  int lane = threadIdx.x & 31;
  int lm = lane & 15, half = lane >> 4;
  int row0 = wid * 16;
  const float* xrow = x + (size_t)(row0 + lm) * DIN;
  v8f acc0 = {}; v8f acc1 = {};
  #pragma unroll
  for (int k0 = 0; k0 < 12; k0 += 4) {        // K=10 padded to 12
    int ka = k0 + half * 2;
    float km = (ka < DIN) ? 1.f : 0.f;        // even pad: one mask per pair
    int kc = (ka < DIN) ? ka : (DIN - 2);     // clamped, even -> 8B aligned
    float2 av = *(const float2*)(xrow + kc);
    v2f a, bf0, bf1;
    a.x = (av.x + PO_SIN[kc]) * km;
    a.y = (av.y + PO_SIN[kc + 1]) * km;
    bf0.x = W1[kc * DH + lm] * km;
    bf0.y = W1[(kc + 1) * DH + lm] * km;
    int c1 = 16 + lm;
    float cm = (c1 < DH) ? km : 0.f;
    int c1c = (c1 < DH) ? c1 : (DH - 1);
    bf1.x = W1[kc * DH + c1c] * cm;
    bf1.y = W1[(kc + 1) * DH + c1c] * cm;
    acc0 = wmma4(a, bf0, acc0);
    acc1 = wmma4(a, bf1, acc1);
  }
  #pragma unroll
  for (int i = 0; i < 8; i++) {
    int m = row0 + i + half * 8;
    float di = dinv[m];
    float d2 = di * di;
    h1pre[(size_t)m * DH + lm] = acc0[i];
    agg[(size_t)m * DH + lm] = acc0[i] * d2;
    if (16 + lm < DH) {
      h1pre[(size_t)m * DH + 16 + lm] = acc1[i];
      agg[(size_t)m * DH + 16 + lm] = acc1[i] * d2;
    }
  }
}

// ---------------- layer-1 edge aggregation ----------------
__global__ void k_edge1(const int* __restrict__ src, const int* __restrict__ dst,
                        const float* __restrict__ dinv,
                        const float* __restrict__ h1pre, float* __restrict__ agg) {
  int e = blockIdx.x * blockDim.x + threadIdx.x;
  if (e >= EE) return;
  int s = src[e], d = dst[e];
  float nrm = dinv[s] * dinv[d];
  const float4* hs = (const float4*)(h1pre + (size_t)s * DH); // 80B rows -> 16B aligned
  float* ad = agg + (size_t)d * DH;
  #pragma unroll
  for (int q = 0; q < 5; q++) {
    float4 v = hs[q];
    atomicAdd(&ad[q * 4 + 0], v.x * nrm);
    atomicAdd(&ad[q * 4 + 1], v.y * nrm);
    atomicAdd(&ad[q * 4 + 2], v.z * nrm);
    atomicAdd(&ad[q * 4 + 3], v.w * nrm);
  }
}

__global__ void k_post1(float* agg, const float* __restrict__ b1) {
  int i = blockIdx.x * blockDim.x + threadIdx.x;
  if (i >= NN * DH) return;
  agg[i] = gelu_f(agg[i] + b1[i % DH]);
}

// ---- h2pre = h1 @ W2  [N,20]x[20,5]; also agg = h2pre * dinv^2 ----
__global__ void k_gemm_w2(const float* __restrict__ h1, const float* __restrict__ W2,
                          const float* __restrict__ dinv,
                          float* __restrict__ h2pre, float* __restrict__ agg) {
  int wid = (int)((blockIdx.x * blockDim.x + threadIdx.x) >> 5);
  if (wid >= NN / 16) return;
  int lane = threadIdx.x & 31;
  int lm = lane & 15, half = lane >> 4;
  int row0 = wid * 16;
  const float* hrow = h1 + (size_t)(row0 + lm) * DH;
  float cm = (lm < DE) ? 1.f : 0.f;
  int lmc = (lm < DE) ? lm : (DE - 1);
  v8f acc = {};
  #pragma unroll
  for (int k0 = 0; k0 < DH; k0 += 4) {        // K=20, no pad
    int ka = k0 + half * 2;
    float2 av = *(const float2*)(hrow + ka);  // ka even -> 8B aligned
    v2f a, b;
    a.x = av.x; a.y = av.y;
    b.x = W2[ka * DE + lmc] * cm;
    b.y = W2[(ka + 1) * DE + lmc] * cm;
    acc = wmma4(a, b, acc);
  }
  if (lm < DE) {
    #pragma unroll
    for (int i = 0; i < 8; i++) {
      int m = row0 + i + half * 8;
      float di = dinv[m];
      h2pre[(size_t)m * DE + lm] = acc[i];
      agg[(size_t)m * DE + lm] = acc[i] * di * di;
    }
  }
}

// ---------------- layer-2 edge aggregation ----------------
__global__ void k_edge2(const int* __restrict__ src, const int* __restrict__ dst,
                        const float* __restrict__ dinv,
                        const float* __restrict__ h2pre, float* __restrict__ agg) {
  int e = blockIdx.x * blockDim.x + threadIdx.x;
  if (e >= EE) return;
  int s = src[e], d = dst[e];
  float nrm = dinv[s] * dinv[d];
  const float* hs = h2pre + (size_t)s * DE;
  float* ad = agg + (size_t)d * DE;
  #pragma unroll
  for (int c = 0; c < DE; c++) atomicAdd(&ad[c], hs[c] * nrm);
}

__global__ void k_post2(float* agg, const float* __restrict__ b2) {
  int i = blockIdx.x * blockDim.x + threadIdx.x;
  if (i >= NN * DE) return;
  agg[i] = gelu_f(agg[i] + b2[i % DE]);
}

// ---------------- fused tail: enc / out / x1 ----------------
// r [B,30] -> enc = r@We+be ; out = gelu(enc)@Wd+bd ; x1 = enc@Wr+br
__global__ void k_tail(const float* __restrict__ r,
                       const float* __restrict__ We, const float* __restrict__ be,
                       const float* __restrict__ Wd, const float* __restrict__ bd,
                       const float* __restrict__ Wr, const float* __restrict__ br,
                       float* __restrict__ x1, float* __restrict__ enc,
                       float* __restrict__ outp) {
  __shared__ float lds_enc[8][16][32];   // zero-padded to K=32
  __shared__ float lds_gel[8][16][32];
  int w = threadIdx.x >> 5;
  int wid = blockIdx.x * 8 + w;
  bool active = wid < BB / 16;           // wave-uniform
  int lane = threadIdx.x & 31;
  int lm = lane & 15, half = lane >> 4;
  int row0 = wid * 16;

  if (active) {
    const float* rrow = r + (size_t)(row0 + lm) * DR;
    int c1 = 16 + lm;
    float cm1 = (c1 < DR) ? 1.f : 0.f;
    int c1c = (c1 < DR) ? c1 : (DR - 1);
    v8f e0 = {}, e1 = {};
    #pragma unroll
    for (int k0 = 0; k0 < 32; k0 += 4) {     // K=30 padded to 32
      int ka = k0 + half * 2;
      float km = (ka < DR) ? 1.f : 0.f;      // even pad: one mask per pair
      int kc = (ka < DR) ? ka : (DR - 2);    // clamped, even -> 8B aligned
      float2 av = *(const float2*)(rrow + kc);
      v2f a, bf0, bf1;
      a.x = av.x * km;
      a.y = av.y * km;
      bf0.x = We[kc * DR + lm] * km;
      bf0.y = We[(kc + 1) * DR + lm] * km;
      bf1.x = We[kc * DR + c1c] * (km * cm1);
      bf1.y = We[(kc + 1) * DR + c1c] * (km * cm1);
      e0 = wmma4(a, bf0, e0);
      e1 = wmma4(a, bf1, e1);
    }
    #pragma unroll
    for (int i = 0; i < 8; i++) {
      int m = i + half * 8;
      int row = row0 + m;
      float v0 = e0[i] + be[lm];
      enc[(size_t)row * DR + lm] = v0;
      lds_enc[w][m][lm] = v0;
      lds_gel[w][m][lm] = gelu_f(v0);
      if (c1 < DR) {
        float v1 = e1[i] + be[c1];
        enc[(size_t)row * DR + c1] = v1;
        lds_enc[w][m][c1] = v1;
        lds_gel[w][m][c1] = gelu_f(v1);
      } else {
        lds_enc[w][m][c1] = 0.f;
        lds_gel[w][m][c1] = 0.f;
      }
    }
  }
  __syncthreads();
  if (!active) return;

  // out = gelu(enc) @ Wd + bd : 4 col tiles (60 cols, pad to 64)
  #pragma unroll
  for (int ct = 0; ct < 4; ct++) {
    int col = ct * 16 + lm;
    float cmo = (col < DOUT) ? 1.f : 0.f;
    int colc = (col < DOUT) ? col : (DOUT - 1);
    v8f acc = {};
    #pragma unroll
    for (int k0 = 0; k0 < 32; k0 += 4) {
      int ka = k0 + half * 2;
      float km = (ka < DR) ? 1.f : 0.f;
      int kc = (ka < DR) ? ka : (DR - 2);
      v2f a, b;
      a.x = lds_gel[w][lm][ka];              // LDS already zero-padded
      a.y = lds_gel[w][lm][ka + 1];
      b.x = Wd[kc * DOUT + colc] * (km * cmo);
      b.y = Wd[(kc + 1) * DOUT + colc] * (km * cmo);
      acc = wmma4(a, b, acc);
    }
    if (col < DOUT) {
      #pragma unroll
      for (int i = 0; i < 8; i++)
        outp[(size_t)(row0 + i + half * 8) * DOUT + col] = acc[i] + bd[col];
    }
  }

  // x1 = enc @ Wr + br : 1 col tile (7 cols)
  {
    float cmx = (lm < DX1) ? 1.f : 0.f;
    int lmc = (lm < DX1) ? lm : (DX1 - 1);
    v8f acc = {};
    #pragma unroll
    for (int k0 = 0; k0 < 32; k0 += 4) {
      int ka = k0 + half * 2;
      float km = (ka < DR) ? 1.f : 0.f;
      int kc = (ka < DR) ? ka : (DR - 2);
      v2f a, b;
      a.x = lds_enc[w][lm][ka];
      a.y = lds_enc[w][lm][ka + 1];
      b.x = Wr[kc * DX1 + lmc] * (km * cmx);
      b.y = Wr[(kc + 1) * DX1 + lmc] * (km * cmx);
      acc = wmma4(a, b, acc);
    }
    if (lm < DX1) {
      #pragma unroll
      for (int i = 0; i < 8; i++)
        x1[(size_t)(row0 + i + half * 8) * DX1 + lm] = acc[i] + br[lm];
    }
  }
}

extern "C" void kernel_launch(void* const* d_in, const int* in_sizes, int n_in,
                              void* d_out, int out_size, void* d_ws, size_t ws_size,
                              hipStream_t stream) {
  const float* x  = (const float*)d_in[0];
  const int*   ei = (const int*)d_in[1];
  const float* W1 = (const float*)d_in[2];
  const float* b1 = (const float*)d_in[3];
  const float* W2 = (const float*)d_in[4];
  const float* b2 = (const float*)d_in[5];
  const float* We = (const float*)d_in[6];
  const float* be = (const float*)d_in[7];
  const float* Wd = (const float*)d_in[8];
  const float* bd = (const float*)d_in[9];
  const float* Wr = (const float*)d_in[10];
  const float* br = (const float*)d_in[11];
  const int* src = ei;
  const int* dst = ei + EE;

  float* ws    = (float*)d_ws;
  float* dinv  = ws;                          // NN
  float* h1pre = ws + (size_t)NN;             // NN*20
  float* h1    = ws + (size_t)NN * 21;        // NN*20 (agg1 -> h1)
  float* h2pre = ws + (size_t)NN;             // reuse dead h1pre region (NN*5)
  float* h2    = ws + (size_t)NN * 6;         // NN*5 (agg2 -> h2 == r)

  float* x1  = (float*)d_out;                          // [B,7]
  float* enc = (float*)d_out + (size_t)BB * DX1;       // [B,30]
  float* out = (float*)d_out + (size_t)BB * (DX1 + DR);// [B,60]

  const int T = 256;
  hipLaunchKernelGGL(k_init_deg, dim3((NN + T - 1) / T), dim3(T), 0, stream, dinv);
  hipLaunchKernelGGL(k_degree,   dim3((EE + T - 1) / T), dim3(T), 0, stream, dst, dinv);
  hipLaunchKernelGGL(k_dinv,     dim3((NN + T - 1) / T), dim3(T), 0, stream, dinv);
  hipLaunchKernelGGL(k_gemm_w1,  dim3(NN / 16 / 8), dim3(T), 0, stream, x, W1, dinv, h1pre, h1);
  hipLaunchKernelGGL(k_edge1,    dim3((EE + T - 1) / T), dim3(T), 0, stream, src, dst, dinv, h1pre, h1);
  hipLaunchKernelGGL(k_post1,    dim3((NN * DH + T - 1) / T), dim3(T), 0, stream, h1, b1);
  hipLaunchKernelGGL(k_gemm_w2,  dim3(NN / 16 / 8), dim3(T), 0, stream, h1, W2, dinv, h2pre, h2);
  hipLaunchKernelGGL(k_edge2,    dim3((EE + T - 1) / T), dim3(T), 0, stream, src, dst, dinv, h2pre, h2);
  hipLaunchKernelGGL(k_post2,    dim3((NN * DE + T - 1) / T), dim3(T), 0, stream, h2, b2);
  hipLaunchKernelGGL(k_tail,     dim3((BB / 16 + 7) / 8), dim3(T), 0, stream,
                     h2, We, be, Wd, bd, Wr, br, x1, enc, out);
}